// VirtualTextureModule_51402168599388
// MI455X (gfx1250) — compile-verified
//
#include <hip/hip_runtime.h>

// ---------------------------------------------------------------------------
// Virtual texture: mip pyramid build (2x2 box) + trilinear (linear-mipmap-
// linear) sampling with wrap addressing.
//
// data:       [1,4096,4096,3] f32   (d_in[0])
// texc:       [1,1024,1024,2] f32   (d_in[1])
// texc_deriv: [1,1024,1024,4] f32   (d_in[2])
// out:        [1,1024,1024,3] f32
//
// d_ws holds mip levels 1..8 (16,776,960 floats = 67.1 MB).
//
// Mip build: one wave32 per 16x16 output tile (all 3 channels).
//  - Tile staging: TENSOR_LOAD_TO_LDS (TDM, 2D D#: 96x32 f32 elements, row
//    stride inDim*3), LDS padding 1 dword per 32 -> row stride 99 dwords
//    (odd mod 64 banks => conflict-free column reads).
//  - Box filter as Out = 0.25 * P*In*P^T; horizontal pair-sum H = In*P^T via
//    chained V_WMMA_F32_16X16X4_F32 (constant pairing matrix P), vertical
//    pair-sum in VALU. Full f32 precision, free under the bandwidth bound.
//  - b96 packed RGB stores.
// ---------------------------------------------------------------------------

#define BASE_DIM 4096
#define OUT_DIM  1024
#define MAX_MIP  8
#define MK_WAVES 4           // waves (tiles) per block in the mip kernel

typedef __attribute__((ext_vector_type(2))) float        v2f;
typedef __attribute__((ext_vector_type(8))) float        v8f;
typedef __attribute__((ext_vector_type(4))) unsigned int v4u;
typedef __attribute__((ext_vector_type(4))) int          v4i;
typedef __attribute__((ext_vector_type(8))) int          v8i;

struct f3 { float x, y, z; };   // 12B packed texel -> global_load/store_b96

// Float offsets of mip level l (1..8) inside d_ws.
__device__ __constant__ unsigned int MIP_OFF_DEV[9] = {
    0u, 0u, 12582912u, 15728640u, 16515072u, 16711680u, 16760832u, 16773120u, 16776192u};
static const unsigned int MIP_OFF_HOST[9] = {
    0u, 0u, 12582912u, 15728640u, 16515072u, 16711680u, 16760832u, 16773120u, 16776192u};

__device__ __forceinline__ v8f wmma_k4(v2f a, v2f b, v8f c) {
  // (neg_a, A, neg_b, B, c_mod, C, reuse_a, reuse_b)
  return __builtin_amdgcn_wmma_f32_16x16x4_f32(false, a, false, b, (short)0, c,
                                               false, false);
}

// Padded LDS index of sequential element n = r*96 + x (pad 1 dword per 32).
__device__ __forceinline__ int lds_idx(int r, int x) {
  return r * 99 + x + (x >> 5);
}
#define SIN_WORDS 3168   // max index 31*99 + 95 + 2 = 3166

// ---------------------------------------------------------------------------
// Mip downsample: one wave -> one 16x16 RGB output tile from 32x32 input tile.
// ---------------------------------------------------------------------------
__global__ void __launch_bounds__(32 * MK_WAVES)
mip_downsample_wmma(const float* __restrict__ src, float* __restrict__ dst,
                    int inDim, int outDim, int numUnits) {
  __shared__ float sIn[MK_WAVES][SIN_WORDS];
  __shared__ float sH[MK_WAVES][32][17];

  const int lane = threadIdx.x & 31;
  const int wv   = threadIdx.x >> 5;
  const int unit = blockIdx.x * MK_WAVES + wv;   // wave-uniform
  if (unit >= numUnits) return;                  // uniform: EXEC stays all-ones

  const int tilesX = outDim >> 4;
  const int tx = unit % tilesX;
  const int ty = unit / tilesX;
  const int ix0 = tx * 32, iy0 = ty * 32;

  float* in      = sIn[wv];
  float (*h)[17] = sH[wv];

#if __has_builtin(__builtin_amdgcn_tensor_load_to_lds)
  {
    // --- Tensor Data Mover: DMA the 32-row x 384-byte tile into LDS -------
    const unsigned lds_off = (unsigned)(size_t)(void*)in;  // low 32 bits = LDS offset
    const unsigned long long ga =
        (unsigned long long)(size_t)(const void*)(src + ((size_t)iy0 * (size_t)inDim + (size_t)ix0) * 3u);

    v4u g0;                                   // D# group 0 (§8.3)
    g0[0] = 1u;                               // count=1, user mode, no gather
    g0[1] = lds_off;                          // lds_addr (bytes)
    g0[2] = (unsigned)(ga & 0xFFFFFFFFu);     // global_addr[31:0]
    g0[3] = (unsigned)((ga >> 32) & 0x1FFFFFFu) | 0x80000000u;  // addr[56:32], type=2

    v8i g1;                                   // D# group 1 (§8.4)
    g1[0] = (int)((2u << 16)    // data_size = 4B
                | (1u << 20)    // pad_enable
                | (4u << 22));  // pad_interval code 4 = 32 dwords (pad_amount code 0 = 1 dword)
    g1[1] = (int)(96u << 16);   // tensor_dim0[15:0] = 96 (atomic_barrier_addr = 0)
    g1[2] = (int)(32u << 16);   // tensor_dim0 hi = 0 ; tensor_dim1[15:0] = 32
    g1[3] = (int)(96u << 16);   // tensor_dim1 hi = 0 ; tile_dim0 = 96
    g1[4] = 32;                 // tile_dim1 = 32 ; tile_dim2 = 0
    g1[5] = inDim * 3;          // tensor_dim0_stride[31:0] (elements)
    g1[6] = 0;                  // stride hi / tensor_dim1_stride lo
    g1[7] = 0;

    v4i gz4 = {0, 0, 0, 0};              // groups 2/3 unused (2D tensor)
    v8i gz8 = {0, 0, 0, 0, 0, 0, 0, 0};  // extra group (clang-23 6-arg form)
    __builtin_amdgcn_tensor_load_to_lds(g0, g1, gz4, gz4, gz8, 0);
    __builtin_amdgcn_s_wait_tensorcnt(0);
    asm volatile("" ::: "memory");
  }
#else
  {
    // Fallback: per-lane b96 texel loads into the same padded LDS layout.
    const f3* s3 = (const f3*)src;
    #pragma unroll 4
    for (int r = 0; r < 32; ++r) {
      f3 p = s3[(size_t)(iy0 + r) * (size_t)inDim + (size_t)(ix0 + lane)];
      const int x = 3 * lane;
      in[lds_idx(r, x)]     = p.x;
      in[lds_idx(r, x + 1)] = p.y;
      in[lds_idx(r, x + 2)] = p.z;
    }
    asm volatile("" ::: "memory");
  }
#endif

  const int hl = lane >> 4;   // lane half: K slots {0,1} vs {2,3}
  const int m  = lane & 15;   // A row / B col / D col

  float outv[3][8];

  #pragma unroll
  for (int ch = 0; ch < 3; ++ch) {
    v8f accT = {0.f,0.f,0.f,0.f,0.f,0.f,0.f,0.f};  // H rows 0..15
    v8f accB = {0.f,0.f,0.f,0.f,0.f,0.f,0.f,0.f};  // H rows 16..31

    #pragma unroll
    for (int k = 0; k < 4; ++k) {
      const int c0 = 4 * k + 2 * hl;                         // column index in 32x32 plane
      const float b0v = (m == (2 * k + hl))     ? 1.0f : 0.0f;  // P0^T fragment
      const float b1v = (m == (2 * k + hl + 8)) ? 1.0f : 0.0f;  // P1^T fragment
      v2f bP0; bP0[0] = b0v; bP0[1] = b0v;
      v2f bP1; bP1[0] = b1v; bP1[1] = b1v;

      v2f a;
      a[0] = in[lds_idx(m, 3 * c0 + ch)];
      a[1] = in[lds_idx(m, 3 * (c0 + 1) + ch)];
      accT = wmma_k4(a, bP0, accT);                          // In00 * P0^T
      a[0] = in[lds_idx(m, 3 * (16 + c0) + ch)];
      a[1] = in[lds_idx(m, 3 * (17 + c0) + ch)];
      accT = wmma_k4(a, bP1, accT);                          // In01 * P1^T
      a[0] = in[lds_idx(16 + m, 3 * c0 + ch)];
      a[1] = in[lds_idx(16 + m, 3 * (c0 + 1) + ch)];
      accB = wmma_k4(a, bP0, accB);                          // In10 * P0^T
      a[0] = in[lds_idx(16 + m, 3 * (16 + c0) + ch)];
      a[1] = in[lds_idx(16 + m, 3 * (17 + c0) + ch)];
      accB = wmma_k4(a, bP1, accB);                          // In11 * P1^T
    }

    // D layout -> row-major H in LDS: VGPR v holds (row v + 8*hl, col m)
    asm volatile("" ::: "memory");
    #pragma unroll
    for (int v = 0; v < 8; ++v) {
      h[v + 8 * hl][m]      = accT[v];
      h[16 + v + 8 * hl][m] = accB[v];
    }
    asm volatile("" ::: "memory");

    // Vertical pair-sum + scale
    #pragma unroll
    for (int r = 0; r < 8; ++r) {
      const int i = r + 8 * hl;
      outv[ch][r] = 0.25f * (h[2 * i][m] + h[2 * i + 1][m]);
    }
    asm volatile("" ::: "memory");   // sH reused by next channel (DS in-order per wave)
  }

  // Packed RGB b96 stores
  f3* dst3 = (f3*)dst;
  const int ox0 = tx * 16, oy0 = ty * 16;
  #pragma unroll
  for (int r = 0; r < 8; ++r) {
    const int i = r + 8 * hl;
    f3 p; p.x = outv[0][r]; p.y = outv[1][r]; p.z = outv[2][r];
    dst3[(size_t)(oy0 + i) * (size_t)outDim + (size_t)(ox0 + m)] = p;
  }
}

// ---------------------------------------------------------------------------
// Trilinear sampler (b96 texel gathers)
// ---------------------------------------------------------------------------
__device__ __forceinline__ void bilinear_wrap_acc(const float* __restrict__ t,
                                                  int dim, float u, float v,
                                                  float w, float acc[3]) {
  const float x = u * (float)dim - 0.5f;
  const float y = v * (float)dim - 0.5f;
  const float xf = floorf(x), yf = floorf(y);
  const float fx = x - xf,    fy = y - yf;
  const int msk = dim - 1;                       // dims are powers of two
  const int x0 = ((int)xf) & msk,  y0 = ((int)yf) & msk;
  const int x1 = (x0 + 1) & msk,   y1 = (y0 + 1) & msk;
  const float w00 = (1.f - fx) * (1.f - fy) * w;
  const float w10 = fx * (1.f - fy) * w;
  const float w01 = (1.f - fx) * fy * w;
  const float w11 = fx * fy * w;
  const f3* t3 = (const f3*)t;
  const size_t r0 = (size_t)y0 * (size_t)dim;
  const size_t r1 = (size_t)y1 * (size_t)dim;
  f3 p00 = t3[r0 + x0];
  f3 p10 = t3[r0 + x1];
  f3 p01 = t3[r1 + x0];
  f3 p11 = t3[r1 + x1];
  acc[0] += w00 * p00.x + w10 * p10.x + w01 * p01.x + w11 * p11.x;
  acc[1] += w00 * p00.y + w10 * p10.y + w01 * p01.y + w11 * p11.y;
  acc[2] += w00 * p00.z + w10 * p10.z + w01 * p01.z + w11 * p11.z;
}

__global__ void __launch_bounds__(256)
trilinear_sample(const float* __restrict__ data, const float* __restrict__ mips,
                 const float* __restrict__ texc, const float* __restrict__ deriv,
                 float* __restrict__ out) {
  const int idx = blockIdx.x * blockDim.x + threadIdx.x;
  if (idx >= OUT_DIM * OUT_DIM) return;

  const float u = texc[2 * idx + 0];
  const float v = texc[2 * idx + 1];
  const float d0 = deriv[4 * idx + 0] * (float)BASE_DIM;
  const float d1 = deriv[4 * idx + 1] * (float)BASE_DIM;
  const float d2 = deriv[4 * idx + 2] * (float)BASE_DIM;
  const float d3 = deriv[4 * idx + 3] * (float)BASE_DIM;
  const float rho2 = fmaxf(d0 * d0 + d1 * d1, d2 * d2 + d3 * d3);
  float lod = 0.5f * __log2f(fmaxf(rho2, 1e-20f));
  lod = fminf(fmaxf(lod, 0.0f), (float)MAX_MIP);

  int l0 = (int)floorf(lod);
  if (l0 > MAX_MIP - 1) l0 = MAX_MIP - 1;   // lod==8 -> (l0=7, f=1): all weight on level 8
  const float f = lod - (float)l0;
  const int l1 = l0 + 1;

  float acc[3] = {0.f, 0.f, 0.f};
  {
    const int dim = BASE_DIM >> l0;
    const float* t = (l0 == 0) ? data : (mips + MIP_OFF_DEV[l0]);
    bilinear_wrap_acc(t, dim, u, v, 1.0f - f, acc);
  }
  {
    const int dim = BASE_DIM >> l1;
    const float* t = mips + MIP_OFF_DEV[l1];  // l1 >= 1 always
    bilinear_wrap_acc(t, dim, u, v, f, acc);
  }
  out[3 * idx + 0] = acc[0];
  out[3 * idx + 1] = acc[1];
  out[3 * idx + 2] = acc[2];
}

// ---------------------------------------------------------------------------
extern "C" void kernel_launch(void* const* d_in, const int* in_sizes, int n_in,
                              void* d_out, int out_size, void* d_ws, size_t ws_size,
                              hipStream_t stream) {
  (void)in_sizes; (void)n_in; (void)out_size; (void)ws_size;
  const float* data  = (const float*)d_in[0];
  const float* texc  = (const float*)d_in[1];
  const float* deriv = (const float*)d_in[2];
  float* out = (float*)d_out;
  float* ws  = (float*)d_ws;

  // Build mip levels 1..8 (levels >=1 stay L2-resident for the next pass)
  const float* src = data;
  for (int l = 1; l <= MAX_MIP; ++l) {
    const int inDim  = BASE_DIM >> (l - 1);
    const int outDim = BASE_DIM >> l;
    float* dst = ws + MIP_OFF_HOST[l];
    const int tiles = outDim >> 4;
    const int numUnits = tiles * tiles;                       // one RGB tile per wave
    const int blocks = (numUnits + MK_WAVES - 1) / MK_WAVES;
    mip_downsample_wmma<<<blocks, 32 * MK_WAVES, 0, stream>>>(src, dst, inDim, outDim, numUnits);
    src = dst;
  }

  const int n = OUT_DIM * OUT_DIM;
  trilinear_sample<<<(n + 255) / 256, 256, 0, stream>>>(data, ws, texc, deriv, out);
}